// VPNNetwork_84567906058394
// MI455X (gfx1250) — compile-verified
//
#include <hip/hip_runtime.h>

typedef __attribute__((ext_vector_type(2))) float v2f;
typedef __attribute__((ext_vector_type(8))) float v8f;

#define BLK 128            // threads per block = 4 waves = 128 samples
#define PHI_STRIDE 49      // 48 padded to 49: gcd(49,64)=1 -> conflict-free
#define V_STRIDE 17        // 16 padded to 17

__global__ __launch_bounds__(BLK)
void vpn_fused_kernel(const float* __restrict__ obs,     // (B,48)
                      const float* __restrict__ Wphi,    // (48,48) row-major [out][in]
                      const float* __restrict__ bphi,    // (48)
                      const float* __restrict__ Wlog,    // (4,36)
                      const float* __restrict__ blog,    // (4)
                      float* __restrict__ out)           // (B,4)
{
    __shared__ float phi_lds[BLK * PHI_STRIDE];
    __shared__ float v_lds[BLK * V_STRIDE];

    const int tid   = threadIdx.x;
    const int wave  = tid >> 5;
    const int lane  = tid & 31;
    const int lo    = lane & 15;   // column / row-in-halfwave index
    const int khalf = lane >> 4;   // which K/M half this lane covers
    const long blockBase = (long)blockIdx.x * BLK;

    // ---------------- Phase 1: phi = obs @ Wphi^T + bphi  (WMMA f32 16x16x4) ---
    // Each wave: two 16-sample row tiles x three 16-feature col tiles, K=48.
    for (int tile = 0; tile < 2; ++tile) {
        const long rowBase = blockBase + wave * 32 + tile * 16;

        v8f acc[3];
        #pragma unroll
        for (int ct = 0; ct < 3; ++ct) {
            float bv = bphi[ct * 16 + lo];     // N = ct*16 + lo, same for all 8 rows
            #pragma unroll
            for (int r = 0; r < 8; ++r) acc[ct][r] = bv;
        }

        // A layout (16x4 f32): lane<16 -> M=lo, {K0,K1}; lane>=16 -> M=lo, {K2,K3}
        const float* aptr = obs + (rowBase + lo) * 48 + khalf * 2;

        #pragma unroll
        for (int kk = 0; kk < 12; ++kk) {
            const int k0 = kk * 4;
            v2f a = *(const v2f*)(aptr + k0);
            #pragma unroll
            for (int ct = 0; ct < 3; ++ct) {
                // B layout (4x16 f32): B[k][n] = Wphi[n*48 + k]; lane holds
                // n = ct*16+lo, k = k0 + khalf*2 + {0,1} -> contiguous pair.
                v2f b = *(const v2f*)(Wphi + (ct * 16 + lo) * 48 + k0 + khalf * 2);
                acc[ct] = __builtin_amdgcn_wmma_f32_16x16x4_f32(
                    false, a, false, b, (short)0, acc[ct], false, false);
            }
        }

        // D layout: lane, VGPR r -> D[M = r + khalf*8][N = lo]
        #pragma unroll
        for (int ct = 0; ct < 3; ++ct) {
            #pragma unroll
            for (int r = 0; r < 8; ++r) {
                int m = r + khalf * 8;
                phi_lds[(wave * 32 + tile * 16 + m) * PHI_STRIDE + ct * 16 + lo] = acc[ct][r];
            }
        }
    }
    __syncthreads();

    // ---------------- Phase 2: per-sample value iteration + policy head -------
    const long s = blockBase + tid;
    const float* ph = &phi_lds[tid * PHI_STRIDE];

    float rin[16], rout[16], pch[16];
    #pragma unroll
    for (int c = 0; c < 16; ++c) {          // phi reshaped (4,4,3): idx = cell*3 + ch
        rin[c]  = ph[c * 3 + 0];
        rout[c] = ph[c * 3 + 1];
        pch[c]  = ph[c * 3 + 2];
    }

    float v[16];
    #pragma unroll
    for (int c = 0; c < 16; ++c) v[c] = 0.f;

    for (int it = 0; it < 20; ++it) {       // K = 20 Bellman steps
        float nv[16];
        #pragma unroll
        for (int h = 0; h < 4; ++h) {
            #pragma unroll
            for (int w = 0; w < 4; ++w) {
                const int c = h * 4 + w;
                const float pc = pch[c], ro = rout[c];
                float best = v[c];
                // west (h, w-1)
                { float vn = (w > 0) ? v[c - 1]   : 0.f;
                  float rn = (w > 0) ? rin[c - 1] : 0.f;
                  best = fmaxf(best, fmaf(pc, vn, rn) - ro); }
                // east (h, w+1)
                { float vn = (w < 3) ? v[c + 1]   : 0.f;
                  float rn = (w < 3) ? rin[c + 1] : 0.f;
                  best = fmaxf(best, fmaf(pc, vn, rn) - ro); }
                // north (h-1, w)
                { float vn = (h > 0) ? v[c - 4]   : 0.f;
                  float rn = (h > 0) ? rin[c - 4] : 0.f;
                  best = fmaxf(best, fmaf(pc, vn, rn) - ro); }
                // south (h+1, w)
                { float vn = (h < 3) ? v[c + 4]   : 0.f;
                  float rn = (h < 3) ? rin[c + 4] : 0.f;
                  best = fmaxf(best, fmaf(pc, vn, rn) - ro); }
                nv[c] = best;
            }
        }
        #pragma unroll
        for (int c = 0; c < 16; ++c) v[c] = nv[c];
    }

    // V = swapaxes(v,1,2): V[h][w] = v[w][h]; put in LDS for dynamic window index
    #pragma unroll
    for (int h = 0; h < 4; ++h)
        #pragma unroll
        for (int w = 0; w < 4; ++w)
            v_lds[tid * V_STRIDE + h * 4 + w] = v[w * 4 + h];

    // agent position: first nonzero of channel 1 in row-major order, else (1,1)
    const float* os = obs + s * 48;
    int idx = -1;
    #pragma unroll
    for (int k = 15; k >= 0; --k)
        if (os[k * 3 + 1] != 0.f) idx = k;
    const int ii = (idx >= 0) ? (idx >> 2) : 1;
    const int jj = (idx >= 0) ? (idx & 3)  : 1;

    // 3x3 window (27 grid vals + 9 V vals) -> 4 logits, accumulated on the fly
    float lg0 = blog[0], lg1 = blog[1], lg2 = blog[2], lg3 = blog[3];
    #pragma unroll
    for (int dh = 0; dh < 3; ++dh) {
        #pragma unroll
        for (int dw = 0; dw < 3; ++dw) {
            const int gh = ii - 1 + dh, gw = jj - 1 + dw;
            const bool inb = (gh >= 0) & (gh < 4) & (gw >= 0) & (gw < 4);
            const int cell = gh * 4 + gw;
            #pragma unroll
            for (int c = 0; c < 3; ++c) {
                const int t = (dh * 3 + dw) * 3 + c;
                float val = 0.f;
                if (inb) val = os[cell * 3 + c];
                lg0 = fmaf(Wlog[0 * 36 + t], val, lg0);
                lg1 = fmaf(Wlog[1 * 36 + t], val, lg1);
                lg2 = fmaf(Wlog[2 * 36 + t], val, lg2);
                lg3 = fmaf(Wlog[3 * 36 + t], val, lg3);
            }
            {
                const int t = 27 + dh * 3 + dw;
                float val = 0.f;
                if (inb) val = v_lds[tid * V_STRIDE + cell];
                lg0 = fmaf(Wlog[0 * 36 + t], val, lg0);
                lg1 = fmaf(Wlog[1 * 36 + t], val, lg1);
                lg2 = fmaf(Wlog[2 * 36 + t], val, lg2);
                lg3 = fmaf(Wlog[3 * 36 + t], val, lg3);
            }
        }
    }

    out[s * 4 + 0] = lg0;
    out[s * 4 + 1] = lg1;
    out[s * 4 + 2] = lg2;
    out[s * 4 + 3] = lg3;
}

extern "C" void kernel_launch(void* const* d_in, const int* in_sizes, int n_in,
                              void* d_out, int out_size, void* d_ws, size_t ws_size,
                              hipStream_t stream) {
    const float* obs  = (const float*)d_in[0];   // (B, 48)
    const float* Wphi = (const float*)d_in[1];   // (48, 48)
    const float* bphi = (const float*)d_in[2];   // (48,)
    const float* Wlog = (const float*)d_in[3];   // (4, 36)
    const float* blog = (const float*)d_in[4];   // (4,)
    float* out = (float*)d_out;                  // (B, 4)

    const int Bsz = in_sizes[0] / 48;            // 262144
    const int blocks = Bsz / BLK;                // 2048 blocks of 128 samples
    vpn_fused_kernel<<<blocks, BLK, 0, stream>>>(obs, Wphi, bphi, Wlog, blog, out);
}